// DBN_2413771620684
// MI455X (gfx1250) — compile-verified
//
#include <hip/hip_runtime.h>

#define G 256
#define MROWS 262144              // N*F/G = 65536*1024/256
#define INV_M (1.0f/262144.0f)
#define EPS_DBN 1e-8f
#define NS_ITERS 16
#define SR 16                     // slab rows per gram stage buffer
#define GRAM_WGS 256
#define CH 16                     // wm^T rows per apply stage buffer

typedef __attribute__((ext_vector_type(2))) float v2f;
typedef __attribute__((ext_vector_type(8))) float v8f;

__device__ __forceinline__ v8f wmma_f32_16x16x4(v2f a, v2f b, v8f c) {
  // D = A(16x4) x B(4x16) + C, fp32 in/out
  return __builtin_amdgcn_wmma_f32_16x16x4_f32(false, a, false, b, (short)0, c, false, false);
}

// Async global->LDS copy of one 16B packet per call (CDNA5 ASYNCcnt path).
// lds_off: wave-relative LDS byte offset (low 32 bits of flat shared address;
// LDS aperture occupies bits [63:32] of the flat address).
__device__ __forceinline__ void async_b128(unsigned lds_off, const void* gaddr) {
  asm volatile("global_load_async_to_lds_b128 %0, %1, off"
               :: "v"(lds_off), "v"((unsigned long long)gaddr)
               : "memory");
}
__device__ __forceinline__ void wait_async0() {
  asm volatile("s_wait_asynccnt 0x0" ::: "memory");
}

// ---------------------------------------------------------------------------
// Pass 1: gram = A^T A (A = x viewed as [262144,256] row-major) + column sums
// Double-buffered async global->LDS staging, 16 waves, wave w owns i-tile w.
// ---------------------------------------------------------------------------
__global__ __launch_bounds__(512) void k_gram(const float* __restrict__ X,
                                              float* __restrict__ gram,
                                              float* __restrict__ colsum) {
  __shared__ __align__(16) float buf[2][SR * G];   // 2 x 16 KB
  const int tid  = threadIdx.x;
  const int wave = tid >> 5;              // 0..15
  const int lane = tid & 31;
  const int l16  = lane & 15;
  const int hi   = lane >> 4;             // which K-half this lane holds
  const int ibase = wave * 16;            // this wave's i-tile (rows of gram)

  const unsigned lbase[2] = { (unsigned)(size_t)(&buf[0][0]),
                              (unsigned)(size_t)(&buf[1][0]) };

  v8f acc[16] = {};
  float cs[16];
  #pragma unroll
  for (int j = 0; j < 16; ++j) cs[j] = 0.f;

  const int nslabs = MROWS / SR;          // 16384
  const int stride = gridDim.x;
  int s = blockIdx.x;
  int cur = 0;

  // stage 0
  #pragma unroll
  for (int i = 0; i < (SR * G) / (4 * 512); ++i) {   // 2 b128 packets/thread
    const int e = tid + i * 512;                      // b128 packet index
    async_b128(lbase[0] + e * 16, X + (size_t)s * SR * G + e * 4);
  }
  wait_async0();
  __syncthreads();

  while (s < nslabs) {
    const int snext = s + stride;
    if (snext < nslabs) {
      #pragma unroll
      for (int i = 0; i < (SR * G) / (4 * 512); ++i) {
        const int e = tid + i * 512;
        async_b128(lbase[cur ^ 1] + e * 16, X + (size_t)snext * SR * G + e * 4);
      }
    }
    const float* sl = buf[cur];
    #pragma unroll
    for (int k = 0; k < SR; k += 4) {
      const int r0 = k + 2 * hi;
      v2f a;                               // A^T tile: A[M=i, K=r] = X[r, i]
      a.x = sl[(r0    ) * G + ibase + l16];
      a.y = sl[(r0 + 1) * G + ibase + l16];
      #pragma unroll
      for (int j = 0; j < 16; ++j) {
        v2f b;                             // B tile: B[K=r, N=j] = X[r, j]
        b.x = sl[(r0    ) * G + j * 16 + l16];
        b.y = sl[(r0 + 1) * G + j * 16 + l16];
        if (wave == 0) cs[j] += b.x + b.y; // col sums: lanes cover all 4 rows/kstep
        acc[j] = wmma_f32_16x16x4(a, b, acc[j]);
      }
    }
    wait_async0();
    __syncthreads();
    cur ^= 1;
    s = snext;
  }

  // flush partial gram (C/D layout: VGPR v -> M=v (lanes 0-15) / M=v+8 (16-31))
  #pragma unroll
  for (int j = 0; j < 16; ++j) {
    #pragma unroll
    for (int v = 0; v < 8; ++v) {
      const int row = ibase + v + 8 * hi;
      const int col = j * 16 + l16;
      atomicAdd(&gram[row * G + col], acc[j][v]);
    }
  }
  if (wave == 0) {
    #pragma unroll
    for (int j = 0; j < 16; ++j)
      atomicAdd(&colsum[j * 16 + l16], cs[j]);
  }
}

// ---------------------------------------------------------------------------
// Pass 2: sigma = gram/m - mu mu^T + eps*I
// ---------------------------------------------------------------------------
__global__ __launch_bounds__(256) void k_sigma(const float* __restrict__ gram,
                                               const float* __restrict__ colsum,
                                               float* __restrict__ sigma) {
  const int i = blockIdx.x, j = threadIdx.x;
  const float mui = colsum[i] * INV_M;
  const float muj = colsum[j] * INV_M;
  float s = gram[i * G + j] * INV_M - mui * muj;
  if (i == j) s += EPS_DBN;
  sigma[i * G + j] = s;
}

// ---------------------------------------------------------------------------
// Pass 3: Newton-Schulz init: tr = trace(sigma); Y0 = sigma/tr; Z0 = I
// ---------------------------------------------------------------------------
__global__ __launch_bounds__(256) void k_ns_init(const float* __restrict__ sigma,
                                                 float* __restrict__ Y,
                                                 float* __restrict__ Z,
                                                 float* __restrict__ trbuf) {
  __shared__ float red[256];
  const int t = threadIdx.x;
  red[t] = sigma[t * G + t];
  __syncthreads();
  for (int off = 128; off > 0; off >>= 1) {
    if (t < off) red[t] += red[t + off];
    __syncthreads();
  }
  const float tr = red[0];
  if (t == 0) trbuf[0] = tr;
  const float inv = 1.0f / tr;
  for (int idx = t; idx < G * G; idx += 256) {
    Y[idx] = sigma[idx] * inv;
    Z[idx] = ((idx >> 8) == (idx & 255)) ? 1.0f : 0.0f;
  }
}

// ---------------------------------------------------------------------------
// 256x256x256 GEMM: C = alpha*(A@B) + beta*I   (one WMMA wave per 16x16 tile)
// ---------------------------------------------------------------------------
__global__ __launch_bounds__(32) void k_mm256(const float* __restrict__ A,
                                              const float* __restrict__ B,
                                              float* __restrict__ C,
                                              float alpha, float beta) {
  const int it = blockIdx.x * 16, jt = blockIdx.y * 16;
  const int lane = threadIdx.x, l16 = lane & 15, hi = lane >> 4;
  v8f acc = {};
  for (int k = 0; k < G; k += 4) {
    const int r0 = k + 2 * hi;
    v2f a, b;
    a.x = A[(it + l16) * G + r0];
    a.y = A[(it + l16) * G + r0 + 1];
    b.x = B[(r0    ) * G + jt + l16];
    b.y = B[(r0 + 1) * G + jt + l16];
    acc = wmma_f32_16x16x4(a, b, acc);
  }
  #pragma unroll
  for (int v = 0; v < 8; ++v) {
    const int row = it + v + 8 * hi;
    const int col = jt + l16;
    float val = alpha * acc[v];
    if (row == col) val += beta;
    C[row * G + col] = val;
  }
}

// ---------------------------------------------------------------------------
// wm = Z * rsqrt(tr), stored transposed; bvec[g] = -sum_g' wm[g,g'] * mu[g']
// ---------------------------------------------------------------------------
__global__ __launch_bounds__(256) void k_wm(const float* __restrict__ Z,
                                            const float* __restrict__ colsum,
                                            const float* __restrict__ trbuf,
                                            float* __restrict__ WMT,
                                            float* __restrict__ bvec) {
  __shared__ float red[256];
  const int g = blockIdx.x, t = threadIdx.x;
  const float s = rsqrtf(trbuf[0]);
  const float w = Z[g * G + t] * s;
  WMT[t * G + g] = w;                     // transposed for conflict-free B frags
  red[t] = w * (colsum[t] * INV_M);
  __syncthreads();
  for (int off = 128; off > 0; off >>= 1) {
    if (t < off) red[t] += red[t + off];
    __syncthreads();
  }
  if (t == 0) bvec[g] = -red[0];
}

// ---------------------------------------------------------------------------
// Pass 5: OUT[r,g] = (sum_g' X[r,g'] * wm[g,g']) + b[g], then *weight + bias
// wm^T staged through LDS in double-buffered async chunks of CH rows.
// ---------------------------------------------------------------------------
__global__ __launch_bounds__(256) void k_apply(const float* __restrict__ X,
                                               const float* __restrict__ WMT,
                                               const float* __restrict__ weight,
                                               const float* __restrict__ bias,
                                               const float* __restrict__ bvec,
                                               float* __restrict__ out) {
  __shared__ __align__(16) float wch[2][CH * G];   // 2 x 16 KB
  __shared__ float wv[1024];
  __shared__ float bb[1024];
  __shared__ float bv[G];

  const int tid  = threadIdx.x;
  const int wave = tid >> 5;              // 0..7
  const int lane = tid & 31;
  const int l16  = lane & 15;
  const int hi   = lane >> 4;
  const int rbase = (blockIdx.x * 8 + wave) * 16;   // 16-row strip per wave

  const unsigned lbase[2] = { (unsigned)(size_t)(&wch[0][0]),
                              (unsigned)(size_t)(&wch[1][0]) };

  for (int idx = tid; idx < 1024; idx += 256) { wv[idx] = weight[idx]; bb[idx] = bias[idx]; }
  if (tid < G) bv[tid] = bvec[tid];

  // stage chunk 0
  #pragma unroll
  for (int i = 0; i < (CH * G) / (4 * 256); ++i) {  // 4 b128 packets/thread
    const int e = tid + i * 256;
    async_b128(lbase[0] + e * 16, WMT + e * 4);
  }
  wait_async0();
  __syncthreads();

  v8f acc[16] = {};
  int cur = 0;
  for (int kc = 0; kc < G; kc += CH) {
    if (kc + CH < G) {
      #pragma unroll
      for (int i = 0; i < (CH * G) / (4 * 256); ++i) {
        const int e = tid + i * 256;
        async_b128(lbase[cur ^ 1] + e * 16, WMT + (size_t)(kc + CH) * G + e * 4);
      }
    }
    const float* wc = wch[cur];
    #pragma unroll
    for (int k = 0; k < CH; k += 4) {
      const int r0 = k + 2 * hi;
      v2f a;                               // A[M=row, K=g'] = X[row, g']
      a.x = X[(size_t)(rbase + l16) * G + kc + r0];
      a.y = X[(size_t)(rbase + l16) * G + kc + r0 + 1];
      #pragma unroll
      for (int j = 0; j < 16; ++j) {
        v2f b;                             // B[K=g', N=g] = wm[g,g'] = WMT[g',g]
        b.x = wc[(r0    ) * G + j * 16 + l16];
        b.y = wc[(r0 + 1) * G + j * 16 + l16];
        acc[j] = wmma_f32_16x16x4(a, b, acc[j]);
      }
    }
    wait_async0();
    __syncthreads();
    cur ^= 1;
  }

  #pragma unroll
  for (int j = 0; j < 16; ++j) {
    #pragma unroll
    for (int v = 0; v < 8; ++v) {
      const int row = rbase + v + 8 * hi;
      const int col = j * 16 + l16;
      const int f   = ((row & 3) << 8) + col;     // f = (r%4)*256 + g
      const float val = acc[j][v] + bv[col];
      out[(size_t)row * G + col] = val * wv[f] + bb[f];
    }
  }
}

// ---------------------------------------------------------------------------
extern "C" void kernel_launch(void* const* d_in, const int* in_sizes, int n_in,
                              void* d_out, int out_size, void* d_ws, size_t ws_size,
                              hipStream_t stream) {
  (void)in_sizes; (void)n_in; (void)out_size; (void)ws_size;
  const float* X      = (const float*)d_in[0];
  const float* weight = (const float*)d_in[1];
  const float* bias   = (const float*)d_in[2];
  float* out = (float*)d_out;
  float* ws  = (float*)d_ws;

  float* gram   = ws + 0 * 65536;
  float* sigma  = ws + 1 * 65536;
  float* Y      = ws + 2 * 65536;
  float* Z      = ws + 3 * 65536;
  float* Yn     = ws + 4 * 65536;
  float* Zn     = ws + 5 * 65536;
  float* T      = ws + 6 * 65536;
  float* WMT    = ws + 7 * 65536;
  float* colsum = ws + 8 * 65536;
  float* bvec   = colsum + 256;
  float* trbuf  = bvec + 256;

  hipMemsetAsync(gram,   0, 65536 * sizeof(float), stream);
  hipMemsetAsync(colsum, 0,   256 * sizeof(float), stream);

  k_gram<<<GRAM_WGS, 512, 0, stream>>>(X, gram, colsum);
  k_sigma<<<256, 256, 0, stream>>>(gram, colsum, sigma);
  k_ns_init<<<1, 256, 0, stream>>>(sigma, Y, Z, trbuf);
  for (int it = 0; it < NS_ITERS; ++it) {
    k_mm256<<<dim3(16, 16), 32, 0, stream>>>(Z, Y, T, -1.0f, 3.0f);  // T  = 3I - Z@Y
    k_mm256<<<dim3(16, 16), 32, 0, stream>>>(Y, T, Yn, 0.5f, 0.0f);  // Y' = 0.5 Y@T
    k_mm256<<<dim3(16, 16), 32, 0, stream>>>(T, Z, Zn, 0.5f, 0.0f);  // Z' = 0.5 T@Z
    float* tmp;
    tmp = Y; Y = Yn; Yn = tmp;
    tmp = Z; Z = Zn; Zn = tmp;
  }
  k_wm<<<256, 256, 0, stream>>>(Z, colsum, trbuf, WMT, bvec);
  k_apply<<<2048, 256, 0, stream>>>(X, WMT, weight, bias, bvec, out);
}